// KDTreeLayer_70677981823084
// MI455X (gfx1250) — compile-verified
//
#include <hip/hip_runtime.h>

typedef float v2f __attribute__((ext_vector_type(2)));
typedef float v4f __attribute__((ext_vector_type(4)));
typedef float v8f __attribute__((ext_vector_type(8)));

#define N_PTS   8192
#define M_QRY   4096
#define KNN     16
#define CHUNK   256
#define CSTRIDE 20          // padded floats per point-column in LDS (bank spread)
#define BATCH   8           // Phase-B LDS read batch (clause + single wait)

__global__ __launch_bounds__(32)
void knn_wmma_kernel(const float* __restrict__ xyz,
                     const float* __restrict__ new_xyz,
                     float* __restrict__ out)
{
    __shared__ __align__(16) float sdist[CHUNK * CSTRIDE];  // [col][qrow], ~20 KB
    __shared__ float smd[32 * KNN];
    __shared__ int   smi[32 * KNN];

    const int lane = threadIdx.x;      // 0..31
    const int sub  = lane & 15;
    const int hi   = lane >> 4;        // 0: lanes 0-15, 1: lanes 16-31

    const int tile  = blockIdx.x;      // 0..1023
    const int bIdx  = tile >> 8;       // tile / 256
    const int mBase = (tile & 255) << 4;

    // ---- Load A (16 queries x 4 dims), pre-scaled by -2 so C = -2*q.p ----
    // 32-bit A 16x4 layout: VGPR0 = K0 (lanes<16) / K2 (lanes>=16); VGPR1 = K1 / K3
    const float* qp = new_xyz + ((size_t)bIdx * M_QRY + mBase + sub) * 3;
    const float qx = qp[0], qy = qp[1], qz = qp[2];
    v2f a;
    a.x = -2.0f * (hi ? qz : qx);
    a.y = -2.0f * (hi ? 0.0f : qy);

    // per-lane top-16 list, ascending (bd[15] = current worst kept)
    float bd[KNN];
    int   bi[KNN];
#pragma unroll
    for (int j = 0; j < KNN; ++j) { bd[j] = 3.0e38f; bi[j] = 0; }

    const float* pbase = xyz + (size_t)bIdx * N_PTS * 3;
    const int q    = lane >> 1;        // Phase-B query row (2 lanes per query)
    const int half = lane & 1;
    const float* srow = &sdist[q];     // Phase-B base: read srow[col*CSTRIDE]

    for (int cb = 0; cb < N_PTS; cb += CHUNK) {
        // ================= Phase A: WMMA distance tile -> LDS =================
        for (int t = 0; t < CHUNK / 16; ++t) {
            const int col = t * 16 + sub;
            const float* pp = pbase + (size_t)(cb + col) * 3;
            const float px = pp[0], py = pp[1], pz = pp[2];
            v2f b;
            b.x = hi ? pz   : px;
            b.y = hi ? 0.0f : py;
            v8f cz = {};
            // C[m][n] = -2 * q_m . p_n   (16x16 tile)
            v8f c = __builtin_amdgcn_wmma_f32_16x16x4_f32(
                false, a, false, b, (short)0, cz, false, false);
            const float p2 = px * px + py * py + pz * pz;
            // C/D layout: VGPR r -> M=r (lanes<16) or M=r+8 (lanes>=16), N=lane%16
            v4f d0, d1;
            d0.x = c[0] + p2; d0.y = c[1] + p2; d0.z = c[2] + p2; d0.w = c[3] + p2;
            d1.x = c[4] + p2; d1.y = c[5] + p2; d1.z = c[6] + p2; d1.w = c[7] + p2;
            float* dst = &sdist[col * CSTRIDE + hi * 8];
            *(v4f*)(dst)     = d0;
            *(v4f*)(dst + 4) = d1;
        }
        __syncthreads();

        // ================= Phase B: per-lane top-16 maintenance ===============
        const int base = half * (CHUNK / 2);
        for (int i = 0; i < CHUNK / 2; i += BATCH) {
            // batched LDS reads: clause of ds_load_b32 with immediate offsets
            float dv[BATCH];
#pragma unroll
            for (int j = 0; j < BATCH; ++j)
                dv[j] = srow[(base + i + j) * CSTRIDE];

            // batch early-reject: one compare vs current worst
            float bmn = dv[0];
#pragma unroll
            for (int j = 1; j < BATCH; ++j) bmn = fminf(bmn, dv[j]);

            if (bmn < bd[KNN - 1]) {
#pragma unroll
                for (int j = 0; j < BATCH; ++j) {
                    const float d = dv[j];
                    if (d < bd[KNN - 1]) {
                        const int nid = cb + base + i + j;
                        bool lt[KNN];
#pragma unroll
                        for (int k = 0; k < KNN; ++k) lt[k] = d < bd[k];
                        // branchless sorted insert:
                        // new[k] = lt[k] ? (lt[k-1] ? bd[k-1] : d) : bd[k]
#pragma unroll
                        for (int k = KNN - 1; k >= 1; --k) {
                            const float nd = lt[k - 1] ? bd[k - 1] : d;
                            const int   ni = lt[k - 1] ? bi[k - 1] : nid;
                            bd[k] = lt[k] ? nd : bd[k];
                            bi[k] = lt[k] ? ni : bi[k];
                        }
                        if (lt[0]) { bd[0] = d; bi[0] = nid; }
                    }
                }
            }
        }
        __syncthreads();
    }

    // ================= Final merge: 2 sorted 16-lists -> top-16 ===============
#pragma unroll
    for (int j = 0; j < KNN; ++j) {
        smd[lane * KNN + j] = bd[j];
        smi[lane * KNN + j] = bi[j];
    }
    __syncthreads();

    if (lane < 16) {
        const float* da = &smd[(2 * lane) * KNN];
        const int*   ia = &smi[(2 * lane) * KNN];
        const float* db = &smd[(2 * lane + 1) * KNN];
        const int*   ib = &smi[(2 * lane + 1) * KNN];
        float* op = out + ((size_t)bIdx * M_QRY + mBase + lane) * KNN;
        int ai = 0, bj = 0;
        for (int k = 0; k < KNN; ++k) {
            const float va = da[ai], vb = db[bj];
            const bool ta = va <= vb;
            const int sel = ta ? ia[ai] : ib[bj];
            ai += ta ? 1 : 0;
            bj += ta ? 0 : 1;
            op[k] = (float)sel;
        }
    }
}

extern "C" void kernel_launch(void* const* d_in, const int* in_sizes, int n_in,
                              void* d_out, int out_size, void* d_ws, size_t ws_size,
                              hipStream_t stream) {
    (void)in_sizes; (void)n_in; (void)d_ws; (void)ws_size; (void)out_size;
    const float* xyz     = (const float*)d_in[0];   // (4, 8192, 3)
    const float* new_xyz = (const float*)d_in[1];   // (4, 4096, 3)
    float* out = (float*)d_out;                     // (4, 4096, 16) indices as f32

    const int blocks = 4 * (M_QRY / 16);            // 1024 query tiles
    knn_wmma_kernel<<<dim3(blocks), dim3(32), 0, stream>>>(xyz, new_xyz, out);
}